// EdgeToTriAttention_32925219291978
// MI455X (gfx1250) — compile-verified
//
#include <hip/hip_runtime.h>
#include <math.h>

#define N_INC   12288
#define D_K     8
#define NEDGES  6144
#define NTRI    4096

typedef float v2f __attribute__((ext_vector_type(2)));
typedef float v8f __attribute__((ext_vector_type(8)));

__device__ __forceinline__ float gelu_exact(float x) {
  return 0.5f * x * (1.0f + erff(x * 0.70710678118654752f));
}
// monotonic float<->int mapping for atomicMax-based segment max
__device__ __forceinline__ int f2ord(float f) {
  int i = __float_as_int(f);
  return (i >= 0) ? i : (i ^ 0x7fffffff);
}
__device__ __forceinline__ float ord2f(int i) {
  return __int_as_float((i >= 0) ? i : (i ^ 0x7fffffff));
}

// ---------------- q/k/v projection: [N,2] @ [2,8]^T ----------------
__global__ void qkv_kernel(const float* __restrict__ ef,
                           const float* __restrict__ Wq, const float* __restrict__ bq,
                           const float* __restrict__ Wk, const float* __restrict__ bk,
                           const float* __restrict__ Wv, const float* __restrict__ bv,
                           float* __restrict__ q, float* __restrict__ k, float* __restrict__ v) {
  int i = blockIdx.x * blockDim.x + threadIdx.x;
  if (i >= N_INC) return;
  float x0 = ef[2 * i], x1 = ef[2 * i + 1];
#pragma unroll
  for (int o = 0; o < D_K; ++o) {
    q[i * D_K + o] = fmaf(Wq[o * 2], x0, fmaf(Wq[o * 2 + 1], x1, bq[o]));
    k[i * D_K + o] = fmaf(Wk[o * 2], x0, fmaf(Wk[o * 2 + 1], x1, bk[o]));
    v[i * D_K + o] = fmaf(Wv[o * 2], x0, fmaf(Wv[o * 2 + 1], x1, bv[o]));
  }
}

// ---------------- bucket init / count / scan / fill ----------------
__global__ void init_kernel(int* cnt, int* fill, int* segmax, float* segsum) {
  int i = blockIdx.x * blockDim.x + threadIdx.x;
  if (i < NTRI)   { cnt[i] = 0; fill[i] = 0; }
  if (i < NEDGES) { segmax[i] = (int)0x80000000; segsum[i] = 0.0f; }
}

__global__ void count_kernel(const int* __restrict__ tri, int* cnt) {
  int i = blockIdx.x * blockDim.x + threadIdx.x;
  if (i < N_INC) atomicAdd(&cnt[tri[i]], 1);
}

__global__ void scan_kernel(const int* __restrict__ cnt, int* __restrict__ off) {
  __shared__ int partial[256];
  int t = threadIdx.x;
  int base = t * 16;
  int s = 0;
#pragma unroll
  for (int e = 0; e < 16; ++e) s += cnt[base + e];
  partial[t] = s;
  __syncthreads();
  for (int stride = 1; stride < 256; stride <<= 1) {
    int val = (t >= stride) ? partial[t - stride] : 0;
    __syncthreads();
    partial[t] += val;
    __syncthreads();
  }
  int run = partial[t] - s;  // exclusive prefix
#pragma unroll
  for (int e = 0; e < 16; ++e) { off[base + e] = run; run += cnt[base + e]; }
  if (t == 255) off[NTRI] = run;
}

__global__ void fill_kernel(const int* __restrict__ tri, const int* __restrict__ off,
                            int* fill, int* __restrict__ entries) {
  int i = blockIdx.x * blockDim.x + threadIdx.x;
  if (i >= N_INC) return;
  int t = tri[i];
  int pos = off[t] + atomicAdd(&fill[t], 1);
  entries[pos] = i;
}

// ---------------- segmented (triangle-group) attention ----------------
__global__ void attn_kernel(const int* __restrict__ tri, const int* __restrict__ off,
                            const int* __restrict__ entries,
                            const float* __restrict__ q, const float* __restrict__ k,
                            const float* __restrict__ v, float* __restrict__ att) {
  int i = blockIdx.x * blockDim.x + threadIdx.x;
  if (i >= N_INC) return;
  int t = tri[i];
  int beg = off[t], end = off[t + 1];
  const float scale = 0.35355339059327373f;  // 1/sqrt(8)
  float qi[D_K];
#pragma unroll
  for (int d = 0; d < D_K; ++d) qi[d] = q[i * D_K + d];
  float m = -INFINITY;
  for (int p = beg; p < end; ++p) {
    int j = entries[p];
    float s = 0.f;
#pragma unroll
    for (int d = 0; d < D_K; ++d) s = fmaf(qi[d], k[j * D_K + d], s);
    m = fmaxf(m, s * scale);
  }
  float se = 0.f;
  float acc[D_K];
#pragma unroll
  for (int d = 0; d < D_K; ++d) acc[d] = 0.f;
  for (int p = beg; p < end; ++p) {
    int j = entries[p];
    float s = 0.f;
#pragma unroll
    for (int d = 0; d < D_K; ++d) s = fmaf(qi[d], k[j * D_K + d], s);
    float e = expf(s * scale - m);
    se += e;
#pragma unroll
    for (int d = 0; d < D_K; ++d) acc[d] = fmaf(e, v[j * D_K + d], acc[d]);
  }
  float inv = 1.0f / se;  // group always contains i -> se > 0
#pragma unroll
  for (int d = 0; d < D_K; ++d) att[i * D_K + d] = acc[d] * inv;
}

// ---------------- fp32 WMMA GEMM helper (V_WMMA_F32_16X16X4_F32) ----------------
// X: [128 x K] activations in LDS (stride 33), Wt: [K x cols] in LDS (stride 33),
// Y: [128 x cols] output in LDS. 4 waves split 8*nColTiles 16x16 tiles.
__device__ __forceinline__ void gemm_tiles(const float* Xl, const float* Wtl, float* Yl,
                                           const float* __restrict__ bias,
                                           int K, int nColTiles, int lane, int wave) {
  const int LD = 33;
  int half = lane >> 4;      // lanes 16..31 hold the second K/M half
  int m = lane & 15;
  int nTiles = 8 * nColTiles;
  for (int t = wave; t < nTiles; t += 4) {
    int rt = (nColTiles == 2) ? (t >> 1) : t;
    int ct = (nColTiles == 2) ? (t & 1) : 0;
    int row0 = rt * 16;
    int col = ct * 16 + m;
    float bv = bias[col];          // C/D: col = lane%16, same across all 8 VGPRs
    v8f acc;
#pragma unroll
    for (int g = 0; g < 8; ++g) acc[g] = bv;
    for (int kc = 0; kc < K; kc += 4) {
      int ka = kc + half * 2;      // A 16x4: VGPR g <-> K = kc + 2*half + g
      v2f a, b;
      a.x = Xl[(row0 + m) * LD + ka];
      a.y = Xl[(row0 + m) * LD + ka + 1];
      b.x = Wtl[ka * LD + col];    // B 4x16 mirrors A: K striped over VGPR + half
      b.y = Wtl[(ka + 1) * LD + col];
      acc = __builtin_amdgcn_wmma_f32_16x16x4_f32(false, a, false, b, (short)0, acc,
                                                  false, false);
    }
#pragma unroll
    for (int g = 0; g < 8; ++g)    // D 16x16: VGPR g <-> M = g + 8*half
      Yl[(row0 + g + half * 8) * LD + col] = acc[g];
  }
}

// ---------------- fused MLP: 8->32 LN gelu | rms->32 res relu LN | 32->16 gelu | ->1
__global__ void __launch_bounds__(128) mlp_kernel(
    const float* __restrict__ att,
    const float* __restrict__ W1, const float* __restrict__ b1,
    const float* __restrict__ ln1g, const float* __restrict__ ln1b,
    const float* __restrict__ rmsw,
    const float* __restrict__ Wr, const float* __restrict__ br,
    const float* __restrict__ ln2g, const float* __restrict__ ln2b,
    const float* __restrict__ W2, const float* __restrict__ b2,
    const float* __restrict__ W3, const float* __restrict__ b3,
    float* __restrict__ logits) {
  __shared__ float X[128 * 33];
  __shared__ float Y[128 * 33];
  __shared__ float Hs[128 * 33];
  __shared__ float Wt[32 * 33];
  const int LD = 33;
  int tid = threadIdx.x;
  int lane = tid & 31, wave = tid >> 5;
  int rowBase = blockIdx.x * 128;

  for (int idx = tid; idx < 128 * 8; idx += 128) {
    int r = idx >> 3, c = idx & 7;
    X[r * LD + c] = att[(rowBase + r) * 8 + c];
  }
  for (int idx = tid; idx < 256; idx += 128) {        // W1^T  (8 x 32)
    int kk = idx & 7, c = idx >> 3;
    Wt[kk * LD + c] = W1[c * 8 + kk];
  }
  __syncthreads();
  gemm_tiles(X, Wt, Y, b1, 8, 2, lane, wave);         // Y = att @ W1^T + b1
  __syncthreads();
  {  // per-row: LN1 + gelu -> Hs ; rmsnorm(Hs) -> X
    int r = tid;
    float mu = 0.f;
    for (int c = 0; c < 32; ++c) mu += Y[r * LD + c];
    mu *= (1.f / 32.f);
    float var = 0.f;
    for (int c = 0; c < 32; ++c) { float d = Y[r * LD + c] - mu; var = fmaf(d, d, var); }
    var *= (1.f / 32.f);
    float inv = rsqrtf(var + 1e-5f);
    float hrow[32];
    float ss = 0.f;
    for (int c = 0; c < 32; ++c) {
      float xn = (Y[r * LD + c] - mu) * inv * ln1g[c] + ln1b[c];
      float hv = gelu_exact(xn);
      hrow[c] = hv;
      ss = fmaf(hv, hv, ss);
    }
    float rinv = 1.f / (sqrtf(ss * (1.f / 32.f)) + 1e-6f);
    for (int c = 0; c < 32; ++c) {
      Hs[r * LD + c] = hrow[c];
      X[r * LD + c] = hrow[c] * rinv * rmsw[c];
    }
  }
  for (int idx = tid; idx < 1024; idx += 128) {       // Wr^T  (32 x 32)
    int kk = idx & 31, c = idx >> 5;
    Wt[kk * LD + c] = Wr[c * 32 + kk];
  }
  __syncthreads();
  gemm_tiles(X, Wt, Y, br, 32, 2, lane, wave);        // Y = rms(h) @ Wr^T + br
  __syncthreads();
  {  // per-row: h + relu(Y) -> LN2 -> X
    int r = tid;
    float trow[32];
    for (int c = 0; c < 32; ++c) trow[c] = Hs[r * LD + c] + fmaxf(Y[r * LD + c], 0.f);
    float mu = 0.f;
    for (int c = 0; c < 32; ++c) mu += trow[c];
    mu *= (1.f / 32.f);
    float var = 0.f;
    for (int c = 0; c < 32; ++c) { float d = trow[c] - mu; var = fmaf(d, d, var); }
    var *= (1.f / 32.f);
    float inv = rsqrtf(var + 1e-5f);
    for (int c = 0; c < 32; ++c)
      X[r * LD + c] = (trow[c] - mu) * inv * ln2g[c] + ln2b[c];
  }
  for (int idx = tid; idx < 512; idx += 128) {        // W2^T  (32 x 16)
    int kk = idx & 31, c = idx >> 5;
    Wt[kk * LD + c] = W2[c * 32 + kk];
  }
  __syncthreads();
  gemm_tiles(X, Wt, Y, b2, 32, 1, lane, wave);        // Y = h @ W2^T + b2
  __syncthreads();
  {  // per-row: gelu then 16->1
    int r = tid;
    float acc = b3[0];
    for (int c = 0; c < 16; ++c) acc = fmaf(gelu_exact(Y[r * LD + c]), W3[c], acc);
    logits[rowBase + r] = acc;
  }
}

// ---------------- scatter softmax over edge_ids ----------------
__global__ void segmax_kernel(const float* __restrict__ logits, const int* __restrict__ eid,
                              int* segmax) {
  int i = blockIdx.x * blockDim.x + threadIdx.x;
  if (i < N_INC) atomicMax(&segmax[eid[i]], f2ord(logits[i]));
}
__global__ void segexp_kernel(const float* __restrict__ logits, const int* __restrict__ eid,
                              const int* __restrict__ segmax, float* __restrict__ evals,
                              float* segsum) {
  int i = blockIdx.x * blockDim.x + threadIdx.x;
  if (i >= N_INC) return;
  int s = eid[i];
  float e = expf(logits[i] - ord2f(segmax[s]));
  evals[i] = e;
  atomicAdd(&segsum[s], e);
}
__global__ void weights_kernel(const float* __restrict__ evals, const int* __restrict__ eid,
                               const float* __restrict__ segsum, float* __restrict__ out) {
  int i = blockIdx.x * blockDim.x + threadIdx.x;
  if (i < N_INC) out[i] = evals[i] / segsum[eid[i]];
}

extern "C" void kernel_launch(void* const* d_in, const int* in_sizes, int n_in,
                              void* d_out, int out_size, void* d_ws, size_t ws_size,
                              hipStream_t stream) {
  (void)in_sizes; (void)n_in; (void)out_size; (void)ws_size;
  const float* ef   = (const float*)d_in[0];
  const int*   eid  = (const int*)d_in[1];
  const int*   tri  = (const int*)d_in[2];
  const float* Wq = (const float*)d_in[3];  const float* bq = (const float*)d_in[4];
  const float* Wk = (const float*)d_in[5];  const float* bk = (const float*)d_in[6];
  const float* Wv = (const float*)d_in[7];  const float* bv = (const float*)d_in[8];
  const float* W1 = (const float*)d_in[9];  const float* b1 = (const float*)d_in[10];
  const float* ln1g = (const float*)d_in[11]; const float* ln1b = (const float*)d_in[12];
  const float* rmsw = (const float*)d_in[13];
  const float* Wr = (const float*)d_in[14]; const float* br = (const float*)d_in[15];
  const float* ln2g = (const float*)d_in[16]; const float* ln2b = (const float*)d_in[17];
  const float* W2 = (const float*)d_in[18]; const float* b2 = (const float*)d_in[19];
  const float* W3 = (const float*)d_in[20]; const float* b3 = (const float*)d_in[21];
  float* out = (float*)d_out;

  char* p = (char*)d_ws;
  auto carve = [&p](size_t bytes) -> void* {
    void* r = (void*)p;
    p += (bytes + 255) & ~(size_t)255;
    return r;
  };
  float* q      = (float*)carve(N_INC * D_K * sizeof(float));
  float* k      = (float*)carve(N_INC * D_K * sizeof(float));
  float* v      = (float*)carve(N_INC * D_K * sizeof(float));
  float* att    = (float*)carve(N_INC * D_K * sizeof(float));
  float* logits = (float*)carve(N_INC * sizeof(float));
  float* evals  = (float*)carve(N_INC * sizeof(float));
  float* segsum = (float*)carve(NEDGES * sizeof(float));
  int* segmax   = (int*)carve(NEDGES * sizeof(int));
  int* cnt      = (int*)carve(NTRI * sizeof(int));
  int* fill     = (int*)carve(NTRI * sizeof(int));
  int* off      = (int*)carve((NTRI + 1) * sizeof(int));
  int* entries  = (int*)carve(N_INC * sizeof(int));

  dim3 b256(256);
  dim3 gN((N_INC + 255) / 256);
  dim3 gInit((NEDGES + 255) / 256);

  qkv_kernel<<<gN, b256, 0, stream>>>(ef, Wq, bq, Wk, bk, Wv, bv, q, k, v);
  init_kernel<<<gInit, b256, 0, stream>>>(cnt, fill, segmax, segsum);
  count_kernel<<<gN, b256, 0, stream>>>(tri, cnt);
  scan_kernel<<<1, 256, 0, stream>>>(cnt, off);
  fill_kernel<<<gN, b256, 0, stream>>>(tri, off, fill, entries);
  attn_kernel<<<gN, b256, 0, stream>>>(tri, off, entries, q, k, v, att);
  mlp_kernel<<<dim3(N_INC / 128), dim3(128), 0, stream>>>(
      att, W1, b1, ln1g, ln1b, rmsw, Wr, br, ln2g, ln2b, W2, b2, W3, b3, logits);
  segmax_kernel<<<gN, b256, 0, stream>>>(logits, eid, segmax);
  segexp_kernel<<<gN, b256, 0, stream>>>(logits, eid, segmax, evals, segsum);
  weights_kernel<<<gN, b256, 0, stream>>>(evals, eid, segsum, out);
}